// PerChannelQuantizeWrapper_16260746183110
// MI455X (gfx1250) — compile-verified
//
#include <hip/hip_runtime.h>
#include <cstdint>

// Per-row asymmetric fake-quant for a 4096 x 16384 fp32 matrix.
// Memory-bound: one HBM read + one HBM write per element.
// Row (64 KB) is staged into LDS with the gfx1250 async global->LDS engine
// (ASYNCcnt); the min/max pass is pipelined against the in-flight copies via
// partial s_wait_asynccnt waits (async loads retire in order per wave).

typedef float f32x4 __attribute__((ext_vector_type(4)));

constexpr int W      = 16384;      // columns (elements per row)
constexpr int TPB    = 256;        // 8 wave32s per block
constexpr int NV4    = W / 4;      // float4 slots per row = 4096
constexpr int CHUNKS = NV4 / TPB;  // 16 b128 async copies per thread
constexpr int NWAVES = TPB / 32;   // 8
constexpr int CHUNK_BYTES = TPB * 16;  // 4096 B advance per chunk

__global__ __launch_bounds__(TPB) void fq_row_kernel(const float* __restrict__ x,
                                                     float* __restrict__ out) {
  extern __shared__ float srow[];          // W floats = 64 KB dynamic LDS
  __shared__ float smin[NWAVES];
  __shared__ float smax[NWAVES];

  const int    tid     = threadIdx.x;
  const size_t rowbase = (size_t)blockIdx.x * (size_t)W;
  const f32x4* __restrict__ g4 = (const f32x4*)(x + rowbase);
  f32x4*       __restrict__ o4 = (f32x4*)(out + rowbase);
  const f32x4* s4 = (const f32x4*)srow;

  // ---- async stage: whole row -> LDS (CDNA5 ASYNCcnt path, bypasses VGPRs) ----
  // One base LDS-offset VGPR + one base global VGPR pair; INST_OFFSET is added
  // to BOTH the LDS and global addresses (ISA 08 §4.4), and our LDS layout
  // mirrors the row layout, so the 16 copies differ only in offset:
  {
    unsigned int       lds_base = (unsigned int)(uintptr_t)(const void*)(srow) + 16u * tid;
    unsigned long long g_base   = (unsigned long long)(uintptr_t)(g4 + tid);
#pragma unroll
    for (int c = 0; c < CHUNKS; ++c) {
      asm volatile("global_load_async_to_lds_b128 %0, %1, off offset:%2"
                   :: "v"(lds_base), "v"(g_base), "i"(c * CHUNK_BYTES)
                   : "memory");
    }
  }

  // ---- pass 1: min/max, pipelined against the DMA ----
  // Chunk c was this thread's c-th issued async load; loads retire in order,
  // so asynccnt <= (CHUNKS-1-c) guarantees chunk c is in LDS.
  float vmin = 0.0f, vmax = 0.0f;   // seeds fold in min(x_min,0)/max(x_max,0)
#pragma unroll
  for (int c = 0; c < CHUNKS; ++c) {
    asm volatile("s_wait_asynccnt %0" :: "i"(CHUNKS - 1 - c) : "memory");
    f32x4 v = s4[c * TPB + tid];
    vmin = fminf(vmin, fminf(fminf(v.x, v.y), fminf(v.z, v.w)));
    vmax = fmaxf(vmax, fmaxf(fmaxf(v.x, v.y), fmaxf(v.z, v.w)));
  }

  // wave32 butterfly reduction (gfx1250 is wave32-only; width must be 32)
#pragma unroll
  for (int off = 16; off > 0; off >>= 1) {
    vmin = fminf(vmin, __shfl_xor(vmin, off, 32));
    vmax = fmaxf(vmax, __shfl_xor(vmax, off, 32));
  }
  if ((tid & 31) == 0) {
    smin[tid >> 5] = vmin;
    smax[tid >> 5] = vmax;
  }
  __syncthreads();   // every wave has drained asynccnt to 0 by chunk 15

  float rmin = 0.0f, rmax = 0.0f;
#pragma unroll
  for (int wv = 0; wv < NWAVES; ++wv) {
    rmin = fminf(rmin, smin[wv]);
    rmax = fmaxf(rmax, smax[wv]);
  }

  // per-row quantizer parameters (computed redundantly per thread; cheap)
  const float scale = fmaxf((rmax - rmin) / 255.0f, 1e-8f);
  const float zero  = rintf(-rmin / scale);        // round-half-even, as jnp.round
  const float inv   = 1.0f / scale;

  // ---- pass 2: quantize -> clamp -> dequantize, streaming NT stores ----
#pragma unroll
  for (int c = 0; c < CHUNKS; ++c) {
    const int i4 = c * TPB + tid;
    f32x4 v = s4[i4];
    f32x4 r;
    r.x = (fminf(fmaxf(rintf(v.x * inv) + zero, 0.0f), 255.0f) - zero) * scale;
    r.y = (fminf(fmaxf(rintf(v.y * inv) + zero, 0.0f), 255.0f) - zero) * scale;
    r.z = (fminf(fmaxf(rintf(v.z * inv) + zero, 0.0f), 255.0f) - zero) * scale;
    r.w = (fminf(fmaxf(rintf(v.w * inv) + zero, 0.0f), 255.0f) - zero) * scale;
    __builtin_nontemporal_store(r, o4 + i4);       // write-once output: skip cache rinse
  }
}

extern "C" void kernel_launch(void* const* d_in, const int* in_sizes, int n_in,
                              void* d_out, int out_size, void* d_ws, size_t ws_size,
                              hipStream_t stream) {
  const float* x   = (const float*)d_in[0];
  float*       out = (float*)d_out;
  const int    rows = in_sizes[0] / W;   // 4096 for the reference shapes
  fq_row_kernel<<<rows, TPB, W * sizeof(float), stream>>>(x, out);
}